// Mamba3Block_68917045232168
// MI455X (gfx1250) — compile-verified
//
#include <hip/hip_runtime.h>
#include <math.h>

#define B_   2
#define L_   512
#define DM   1024
#define DI   2048
#define DS_  64
#define NH_  32
#define HD_  64
#define NA_  32
#define DIP  4352
#define M1   (B_*L_)

#define NC_  16          // scan chunks
#define LC_  32          // tokens per chunk  (NC_*LC_ == L_)

// section offsets inside one proj row
#define OFF_Z    0
#define OFF_X    2048
#define OFF_B    4096
#define OFF_C    4160
#define OFF_DT   4224
#define OFF_A    4256
#define OFF_TRAP 4288
#define OFF_ANG  4320

typedef __attribute__((ext_vector_type(2))) float v2f;
typedef __attribute__((ext_vector_type(8))) float v8f;

__device__ __forceinline__ float softplus_f(float x) {
    return x > 20.f ? x : log1pf(expf(x));
}
__device__ __forceinline__ float sigmoid_f(float x) {
    return 1.f / (1.f + expf(-x));
}

// ---------------------------------------------------------------------------
// FP32 WMMA GEMM: C[M,N] = A[M,K] @ B[K,N], row-major, lda=K, ldb=ldc=N.
// Requires M % 256 == 0, N % 64 == 0, K % 4 == 0.
// 8 waves/block; each wave computes a 32x64 tile with 8 accumulators
// (B operands amortized over 2 WMMAs, A over 4): 8 wmma per 10 vmem loads.
// ---------------------------------------------------------------------------
__global__ __launch_bounds__(256) void gemm_wmma_f32(
    const float* __restrict__ A, const float* __restrict__ Bm,
    float* __restrict__ C, int M, int N, int K)
{
    const int lane = threadIdx.x & 31;
    const int wave = threadIdx.x >> 5;
    const int lo   = lane & 15;      // row (A) / col (B,C) within tile
    const int hi   = lane >> 4;      // selects K-pair

    const int m0 = blockIdx.y * 256 + wave * 32;
    const int n0 = blockIdx.x * 64;

    const float* Arow0 = A + (size_t)(m0 + lo) * K;
    const float* Arow1 = A + (size_t)(m0 + 16 + lo) * K;

    v8f acc00 = {}, acc01 = {}, acc02 = {}, acc03 = {};
    v8f acc10 = {}, acc11 = {}, acc12 = {}, acc13 = {};

    for (int k = 0; k < K; k += 4) {
        // A 16x4 tiles: a[v] = A[m][k + 2*hi + v]  (contiguous pair -> b64)
        v2f a0; a0.x = Arow0[k + 2*hi]; a0.y = Arow0[k + 2*hi + 1];
        v2f a1; a1.x = Arow1[k + 2*hi]; a1.y = Arow1[k + 2*hi + 1];

        // B 4x16 tiles: b[v] = B[k + 2*hi + v][n0 + lo + 16*j]
        const float* Bp0 = Bm + (size_t)(k + 2*hi) * N + n0 + lo;
        const float* Bp1 = Bp0 + N;
        v2f b0; b0.x = Bp0[0];  b0.y = Bp1[0];
        v2f b1; b1.x = Bp0[16]; b1.y = Bp1[16];
        v2f b2; b2.x = Bp0[32]; b2.y = Bp1[32];
        v2f b3; b3.x = Bp0[48]; b3.y = Bp1[48];

        acc00 = __builtin_amdgcn_wmma_f32_16x16x4_f32(false, a0, false, b0, (short)0, acc00, false, false);
        acc01 = __builtin_amdgcn_wmma_f32_16x16x4_f32(false, a0, false, b1, (short)0, acc01, false, false);
        acc02 = __builtin_amdgcn_wmma_f32_16x16x4_f32(false, a0, false, b2, (short)0, acc02, false, false);
        acc03 = __builtin_amdgcn_wmma_f32_16x16x4_f32(false, a0, false, b3, (short)0, acc03, false, false);
        acc10 = __builtin_amdgcn_wmma_f32_16x16x4_f32(false, a1, false, b0, (short)0, acc10, false, false);
        acc11 = __builtin_amdgcn_wmma_f32_16x16x4_f32(false, a1, false, b1, (short)0, acc11, false, false);
        acc12 = __builtin_amdgcn_wmma_f32_16x16x4_f32(false, a1, false, b2, (short)0, acc12, false, false);
        acc13 = __builtin_amdgcn_wmma_f32_16x16x4_f32(false, a1, false, b3, (short)0, acc13, false, false);
    }

    // C/D layout: VGPR r of lane -> row base + r + 8*hi, col n0 + lo (+16*j)
    #pragma unroll
    for (int r = 0; r < 8; ++r) {
        float* cp0 = C + (size_t)(m0 + r + 8*hi) * N + n0 + lo;
        cp0[0]  = acc00[r];
        cp0[16] = acc01[r];
        cp0[32] = acc02[r];
        cp0[48] = acc03[r];
        float* cp1 = C + (size_t)(m0 + 16 + r + 8*hi) * N + n0 + lo;
        cp1[0]  = acc10[r];
        cp1[16] = acc11[r];
        cp1[32] = acc12[r];
        cp1[48] = acc13[r];
    }
}

// ---------------------------------------------------------------------------
// Per-batch sequential phase cumsum + DT/decay/trap precompute.
// One wave (32 lanes) per batch; lane j = head/angle index.
// ---------------------------------------------------------------------------
__global__ __launch_bounds__(32) void phase_kernel(
    const float* __restrict__ proj, const float* __restrict__ dt_bias,
    float* __restrict__ DTb, float* __restrict__ decayb,
    float* __restrict__ trapb, float* __restrict__ phaseb)
{
    const int b = blockIdx.x;
    const int j = threadIdx.x;                 // 0..31
    float acc = 0.f;
    for (int l = 0; l < L_; ++l) {
        const float* row = proj + (size_t)(b * L_ + l) * DIP;
        float dt = softplus_f(row[OFF_DT + j] + dt_bias[j]);
        float s = dt;                           // wave32 mean over 32 heads
        s += __shfl_xor(s, 16, 32);
        s += __shfl_xor(s,  8, 32);
        s += __shfl_xor(s,  4, 32);
        s += __shfl_xor(s,  2, 32);
        s += __shfl_xor(s,  1, 32);
        const float dtmean = s * (1.f / 32.f);
        acc += row[OFF_ANG + j] * dtmean;

        const size_t o = (size_t)(b * L_ + l) * 32 + j;
        DTb[o]    = dt;
        float Av  = fminf(-softplus_f(row[OFF_A + j]), -1e-4f);
        decayb[o] = expf(Av * dt);
        trapb[o]  = sigmoid_f(row[OFF_TRAP + j]);
        phaseb[o] = acc;
    }
}

// ---------------------------------------------------------------------------
// Per-token layernorm of B/C (64-wide) + RoPE rotation by cumulated phase.
// ---------------------------------------------------------------------------
__global__ __launch_bounds__(64) void rot_kernel(
    const float* __restrict__ proj,
    const float* __restrict__ Bn_g, const float* __restrict__ Bn_b,
    const float* __restrict__ Cn_g, const float* __restrict__ Cn_b,
    const float* __restrict__ phaseb,
    float* __restrict__ Brot, float* __restrict__ Crot)
{
    const int bl = blockIdx.x;
    const int t  = threadIdx.x;                // 0..63
    __shared__ float red[64], nb[64], nc[64];

    const float* row = proj + (size_t)bl * DIP;
    const float vb = row[OFF_B + t];
    const float vc = row[OFF_C + t];

    auto blockReduce = [&](float v) -> float {
        red[t] = v; __syncthreads();
        for (int s = 32; s > 0; s >>= 1) {
            if (t < s) red[t] += red[t + s];
            __syncthreads();
        }
        float r = red[0]; __syncthreads();
        return r;
    };

    const float mB  = blockReduce(vb) * (1.f / 64.f);
    const float vaB = blockReduce((vb - mB) * (vb - mB)) * (1.f / 64.f);
    const float nB  = (vb - mB) * rsqrtf(vaB + 1e-5f) * Bn_g[t] + Bn_b[t];

    const float mC  = blockReduce(vc) * (1.f / 64.f);
    const float vaC = blockReduce((vc - mC) * (vc - mC)) * (1.f / 64.f);
    const float nC  = (vc - mC) * rsqrtf(vaC + 1e-5f) * Cn_g[t] + Cn_b[t];

    nb[t] = nB; nc[t] = nC;
    __syncthreads();

    const int   q  = t >> 1;
    const float ph = phaseb[(size_t)bl * 32 + q];
    const float c  = cosf(ph), s = sinf(ph);
    float ob, oc;
    if ((t & 1) == 0) { ob = nb[t] * c - nb[t + 1] * s; oc = nc[t] * c - nc[t + 1] * s; }
    else              { ob = nb[t - 1] * s + nb[t] * c; oc = nc[t - 1] * s + nc[t] * c; }
    Brot[(size_t)bl * 64 + t] = ob;
    Crot[(size_t)bl * 64 + t] = oc;
}

// ---------------------------------------------------------------------------
// Chunk-parallel SSM scan. Recurrence h_t = d_t*h_{t-1} + inp_t has a SCALAR
// decay per (b,h,t), so it splits into NC_ chunks exactly:
//   h_t = local_h_t + cumdecay_t * H_chunk         (H_chunk = state at entry)
//   y_t = C_t.local_h_t + cumdecay_t * (C_t.H_chunk)
// Bx_prev at a chunk boundary only needs x/Brot of token l0-1 (read from
// global; not recurrence-dependent).
// ---------------------------------------------------------------------------

// Pass A: per-chunk local scan with h0 = 0. One block per (b,h,c).
// Emits raw y (pre-epilogue, into yfin in-place), cumulative decay per token,
// and the chunk's final local state S_c.
__global__ __launch_bounds__(256) void scan_chunk_kernel(
    const float* __restrict__ proj,
    const float* __restrict__ Brot, const float* __restrict__ Crot,
    const float* __restrict__ DTb, const float* __restrict__ decayb,
    const float* __restrict__ trapb,
    float* __restrict__ yfin, float* __restrict__ cumdb,
    float* __restrict__ Sstate)
{
    const int blk = blockIdx.x;                // b*NH_*NC_ + h*NC_ + c
    const int c   = blk % NC_;
    const int h   = (blk / NC_) % NH_;
    const int b   = blk / (NC_ * NH_);
    const int l0  = c * LC_;
    const int t   = threadIdx.x;
    const int p   = t >> 2;                    // 0..63
    const int seg = t & 3;                     // n in [16*seg, 16*seg+16)

    __shared__ float sx[2][64], sB[2][64], sC[2][64], sscal[2][3];

    float hst[16], Bprev[16];
    float xprev = 0.f;
    #pragma unroll
    for (int i = 0; i < 16; ++i) { hst[i] = 0.f; Bprev[i] = 0.f; }
    if (c > 0) {                               // boundary Bx_prev from global
        const size_t blp = (size_t)(b * L_ + l0 - 1);
        xprev = proj[blp * DIP + OFF_X + h * HD_ + p];
        const float* Bp = Brot + blp * 64 + seg * 16;
        #pragma unroll
        for (int i = 0; i < 16; ++i) Bprev[i] = Bp[i];
    }

    auto loadTok = [&](int tok, int buf) {
        const size_t bl = (size_t)(b * L_ + l0 + tok);
        if (t < 64)        sx[buf][t]       = proj[bl * DIP + OFF_X + h * HD_ + t];
        else if (t < 128)  sB[buf][t - 64]  = Brot[bl * 64 + (t - 64)];
        else if (t < 192)  sC[buf][t - 128] = Crot[bl * 64 + (t - 128)];
        else if (t == 192) sscal[buf][0] = DTb[bl * 32 + h];
        else if (t == 193) sscal[buf][1] = decayb[bl * 32 + h];
        else if (t == 194) sscal[buf][2] = trapb[bl * 32 + h];
    };

    loadTok(0, 0);
    __syncthreads();
    float cumd = 1.f;

    for (int tok = 0; tok < LC_; ++tok) {
        const int cur = tok & 1;
        if (tok + 1 < LC_) loadTok(tok + 1, cur ^ 1);   // overlap next loads

        const float x   = sx[cur][p];
        const float dt  = sscal[cur][0];
        const float dec = sscal[cur][1];
        const float tr  = sscal[cur][2];

        float y = 0.f;
        #pragma unroll
        for (int i = 0; i < 16; ++i) {
            const int   n  = seg * 16 + i;
            const float Bn = sB[cur][n];
            const float bx = tr * x * Bn + (1.f - tr) * xprev * Bprev[i];
            hst[i] = dec * hst[i] + dt * bx;
            y += hst[i] * sC[cur][n];
            Bprev[i] = Bn;
        }
        xprev = x;
        cumd *= dec;

        y += __shfl_xor(y, 1, 32);             // 4 lanes sharing p
        y += __shfl_xor(y, 2, 32);

        const size_t bl = (size_t)(b * L_ + l0 + tok);
        if (seg == 0) yfin[bl * DI + h * HD_ + p] = y;               // raw y
        if (t == 0)   cumdb[((size_t)(b * NH_ + h)) * L_ + l0 + tok] = cumd;
        __syncthreads();
    }

    float* Sp = Sstate + (((size_t)(b * NH_ + h)) * NC_ + c) * 4096 + p * 64 + seg * 16;
    #pragma unroll
    for (int i = 0; i < 16; ++i) Sp[i] = hst[i];
}

// Pass B: inter-chunk combine (tiny, NC_ sequential steps). One block per
// (b,h); Sstate is overwritten in place with each chunk's ENTRY state H_c.
__global__ __launch_bounds__(256) void scan_combine_kernel(
    const float* __restrict__ cumdb, float* __restrict__ Sstate)
{
    const int h  = blockIdx.x % NH_;
    const int b  = blockIdx.x / NH_;
    const int t  = threadIdx.x;
    const int p  = t >> 2, seg = t & 3;

    float H[16];
    #pragma unroll
    for (int i = 0; i < 16; ++i) H[i] = 0.f;

    for (int c = 0; c < NC_; ++c) {
        float* Sp = Sstate + (((size_t)(b * NH_ + h)) * NC_ + c) * 4096 + p * 64 + seg * 16;
        const float Pc = cumdb[((size_t)(b * NH_ + h)) * L_ + c * LC_ + LC_ - 1];
        #pragma unroll
        for (int i = 0; i < 16; ++i) {
            const float Sc = Sp[i];
            Sp[i] = H[i];                      // entry state of chunk c
            H[i]  = Pc * H[i] + Sc;
        }
    }
}

// Pass C: fully parallel correction + gating epilogue, in place on yfin.
// y_t += cumdecay_t * (C_t . H_c);  yfin = (y + D*x) * silu(z)
__global__ __launch_bounds__(256) void scan_correct_kernel(
    const float* __restrict__ proj, const float* __restrict__ Crot,
    const float* __restrict__ cumdb, const float* __restrict__ Sstate,
    const float* __restrict__ D_param, float* __restrict__ yfin)
{
    const int blk = blockIdx.x;
    const int c   = blk % NC_;
    const int h   = (blk / NC_) % NH_;
    const int b   = blk / (NC_ * NH_);
    const int l0  = c * LC_;
    const int t   = threadIdx.x;
    const int p   = t >> 2, seg = t & 3;

    __shared__ float sC[64];
    __shared__ float scd;

    float H[16];
    const float* Sp = Sstate + (((size_t)(b * NH_ + h)) * NC_ + c) * 4096 + p * 64 + seg * 16;
    #pragma unroll
    for (int i = 0; i < 16; ++i) H[i] = Sp[i];
    const float Dp = D_param[h];

    for (int tok = 0; tok < LC_; ++tok) {
        const size_t bl = (size_t)(b * L_ + l0 + tok);
        if (t < 64)  sC[t] = Crot[bl * 64 + t];
        if (t == 64) scd   = cumdb[((size_t)(b * NH_ + h)) * L_ + l0 + tok];
        __syncthreads();

        float corr = 0.f;
        #pragma unroll
        for (int i = 0; i < 16; ++i) corr += H[i] * sC[seg * 16 + i];
        corr += __shfl_xor(corr, 1, 32);
        corr += __shfl_xor(corr, 2, 32);

        if (seg == 0) {
            const float x = proj[bl * DIP + OFF_X + h * HD_ + p];
            const float z = proj[bl * DIP + OFF_Z + h * HD_ + p];
            float y = yfin[bl * DI + h * HD_ + p] + scd * corr;
            yfin[bl * DI + h * HD_ + p] = (y + Dp * x) * (z * sigmoid_f(z));
        }
        __syncthreads();
    }
}

// ---------------------------------------------------------------------------
extern "C" void kernel_launch(void* const* d_in, const int* in_sizes, int n_in,
                              void* d_out, int out_size, void* d_ws, size_t ws_size,
                              hipStream_t stream)
{
    const float* u       = (const float*)d_in[0];
    const float* W_in    = (const float*)d_in[1];
    const float* W_out   = (const float*)d_in[2];
    const float* dt_bias = (const float*)d_in[3];
    const float* D_param = (const float*)d_in[4];
    const float* Bn_g    = (const float*)d_in[5];
    const float* Bn_b    = (const float*)d_in[6];
    const float* Cn_g    = (const float*)d_in[7];
    const float* Cn_b    = (const float*)d_in[8];

    float* ws = (float*)d_ws;
    float* proj   = ws; ws += (size_t)M1 * DIP;              // 17.8 MB
    float* yfin   = ws; ws += (size_t)M1 * DI;               //  8.4 MB
    float* Sstate = ws; ws += (size_t)B_ * NH_ * NC_ * 4096; // 16.8 MB
    float* Brot   = ws; ws += (size_t)M1 * DS_;
    float* Crot   = ws; ws += (size_t)M1 * DS_;
    float* DTb    = ws; ws += (size_t)M1 * NH_;
    float* decayb = ws; ws += (size_t)M1 * NH_;
    float* trapb  = ws; ws += (size_t)M1 * NH_;
    float* phaseb = ws; ws += (size_t)M1 * NA_;
    float* cumdb  = ws; ws += (size_t)B_ * NH_ * L_;

    // 1) in-projection GEMM (1024 x 1024 x 4352), fp32 WMMA
    gemm_wmma_f32<<<dim3(DIP / 64, M1 / 256), dim3(256), 0, stream>>>(
        u, W_in, proj, M1, DIP, DM);

    // 2) sequential phase cumsum + DT/decay/trap
    phase_kernel<<<dim3(B_), dim3(32), 0, stream>>>(
        proj, dt_bias, DTb, decayb, trapb, phaseb);

    // 3) layernorm + RoPE of B/C
    rot_kernel<<<dim3(M1), dim3(64), 0, stream>>>(
        proj, Bn_g, Bn_b, Cn_g, Cn_b, phaseb, Brot, Crot);

    // 4) chunk-parallel SSM scan (sequential depth 512 -> 32)
    scan_chunk_kernel<<<dim3(B_ * NH_ * NC_), dim3(256), 0, stream>>>(
        proj, Brot, Crot, DTb, decayb, trapb, yfin, cumdb, Sstate);
    scan_combine_kernel<<<dim3(B_ * NH_), dim3(256), 0, stream>>>(
        cumdb, Sstate);
    scan_correct_kernel<<<dim3(B_ * NH_ * NC_), dim3(256), 0, stream>>>(
        proj, Crot, cumdb, Sstate, D_param, yfin);

    // 5) out-projection GEMM (1024 x 2048 x 1024), fp32 WMMA -> d_out
    gemm_wmma_f32<<<dim3(DM / 64, M1 / 256), dim3(256), 0, stream>>>(
        yfin, W_out, (float*)d_out, M1, DM, DI);
}